// CRF_6597069767072
// MI455X (gfx1250) — compile-verified
//
#include <hip/hip_runtime.h>

#define NTAG   24
#define SEQ    512
#define NBATCH 2048
#define TCH    8                    // time steps per TDM chunk
#define NCHUNK (SEQ / TCH)          // 64
#define WAVES  2                    // waves per block
#define LSTR   NTAG                 // LDS row stride (dwords) per batch
#define SLABDW (16 * LSTR)          // 384 dwords per time-step slab
#define CHDW   (TCH * SLABDW)       // 3072 dwords per chunk buffer

typedef __attribute__((ext_vector_type(2))) float v2f;
typedef __attribute__((ext_vector_type(8))) float v8f;
typedef unsigned int u32;
typedef __attribute__((ext_vector_type(4))) u32 u32x4;
typedef __attribute__((ext_vector_type(8))) int i32x8;
typedef __attribute__((ext_vector_type(4))) int i32x4;

#if defined(__has_builtin)
#if __has_builtin(__builtin_amdgcn_tensor_load_to_lds)
#define HAVE_TDM 1
#endif
#endif

static __device__ __forceinline__ float shf(float v, int src) {
  return __shfl(v, src, 32);
}

#ifdef HAVE_TDM
// DMA one chunk: tile X=24 tags (contig), Y=16 batches (stride SEQ*24),
// Z=TCH timesteps (stride 24) -> LDS contiguous [t][b][j].  D# packed per
// CDNA5 ISA 8.3-8.6.  Tracked on TENSORcnt.  (6-arg clang-23 builtin form;
// 5th int32x8 group passed as zeros.)
static __device__ __forceinline__ void tdm_load_chunk(const float* gsrc, float* ldst) {
  const unsigned long long ga = (unsigned long long)(uintptr_t)gsrc;
  const u32 la = (u32)(uintptr_t)ldst;       // low 32 bits = LDS byte offset
  u32x4 g0;
  g0[0] = 1u;                                              // count=1, user D#
  g0[1] = la;                                              // lds_addr
  g0[2] = (u32)ga;                                         // global_addr[31:0]
  g0[3] = ((u32)(ga >> 32) & 0x01FFFFFFu) | 0x80000000u;   // ga[56:32] | type=2
  i32x8 g1;
  g1[0] = (2 << 16);                 // workgroup_mask=0, data_size=4B, no pad
  g1[1] = (NTAG << 16);              // tensor_dim0[15:0]=24
  g1[2] = (16 << 16);                // tensor_dim0 hi=0 | tensor_dim1[15:0]=16
  g1[3] = (NTAG << 16);              // tensor_dim1 hi=0 | tile_dim0=24
  g1[4] = 16 | (TCH << 16);          // tile_dim1=16 | tile_dim2=TCH
  g1[5] = SEQ * NTAG;                // tensor_dim0_stride = 12288 (batch)
  g1[6] = (NTAG << 16);              // stride0 hi=0 | tensor_dim1_stride=24 (t)
  g1[7] = 0;                         // stride1 hi
  i32x4 g2 = { TCH, 0, 0, 0 };       // tensor_dim2=TCH, no dim3
  i32x4 g3 = { 0, 0, 0, 0 };
  i32x8 gx = { 0, 0, 0, 0, 0, 0, 0, 0 };
  __builtin_amdgcn_tensor_load_to_lds(g0, g1, g2, g3, gx, 0);
}
#else
// Fallback: synchronous per-step staging global -> VGPR -> LDS.
static __device__ __forceinline__ void copy_slab(const float* gsrc, float* ldst, int lane) {
  float ld[12];
#pragma unroll
  for (int k = 0; k < 12; ++k) {
    const int flat = k * 32 + lane;
    ld[k] = gsrc[(size_t)(flat / NTAG) * SEQ * NTAG + (flat % NTAG)];
  }
#pragma unroll
  for (int k = 0; k < 12; ++k) ldst[k * 32 + lane] = ld[k];
}
#endif

// ---------------------------------------------------------------------------
// Log-partition: renormalized linear-space forward scan; one wave = 16 rows.
// raw[j][b] = sum_i expT[i][j] * V[i][b] as 12x v_wmma_f32_16x16x4_f32/step.
// Emissions streamed HBM -> LDS by the Tensor Data Mover, 2 chunks in flight.
// ---------------------------------------------------------------------------
__global__ __launch_bounds__(WAVES * 32)
void crf_partition_kernel(const float* __restrict__ em,
                          const int* __restrict__ mask,
                          const float* __restrict__ T,
                          float* __restrict__ logden)
{
  __shared__ float lds[WAVES * 2 * CHDW];          // 48 KB

  const int  lane = threadIdx.x & 31;
  const int  wave = threadIdx.x >> 5;
  const int  grp  = blockIdx.x * WAVES + wave;     // 0..127
  const int  bcol = lane & 15;
  const bool lo   = lane < 16;

  float* buf0 = &lds[wave * 2 * CHDW];
  float* buf1 = buf0 + CHDW;

  // Constant A operand: A[M=j][K=i] = exp(T[i][j]) in 16x4 f32 A layout.
  v2f A0[6], A1[6];
  {
    const int j0 = bcol;
    const int j1 = (16 + bcol < NTAG) ? (16 + bcol) : (NTAG - 1);
#pragma unroll
    for (int c = 0; c < 6; ++c) {
      const int i0 = 4 * c + (lo ? 0 : 2);
      A0[c].x = __expf(T[(i0    ) * NTAG + j0]);
      A0[c].y = __expf(T[(i0 + 1) * NTAG + j0]);
      A1[c].x = __expf(T[(i0    ) * NTAG + j1]);
      A1[c].y = __expf(T[(i0 + 1) * NTAG + j1]);
    }
  }

  const size_t embase = (size_t)grp * 16 * SEQ * NTAG;
  const int    mrow   = (grp * 16 + bcol) * SEQ;
  const int    off0   = bcol * LSTR + (lo ? 0 : 8);
  const int    off1   = bcol * LSTR + 16;

#ifdef HAVE_TDM
  tdm_load_chunk(em + embase, buf0);                         // chunk 0
  tdm_load_chunk(em + embase + (size_t)TCH * NTAG, buf1);    // chunk 1
  __builtin_amdgcn_s_wait_tensorcnt(1);                      // chunk 0 landed
#else
  copy_slab(em + embase, buf0, lane);
#endif

  // ---- init from t=0 (slab 0 of chunk 0) ----
  float V0[8], V1[8], C;
  {
    float a0[8], a1[8];
#pragma unroll
    for (int r = 0; r < 8; ++r) { a0[r] = buf0[off0 + r]; a1[r] = buf0[off1 + r]; }
    float m0 = a0[0], m1 = a1[0];
#pragma unroll
    for (int r = 1; r < 8; ++r) { m0 = fmaxf(m0, a0[r]); m1 = fmaxf(m1, a1[r]); }
    const float M = fmaxf(fmaxf(m0, __shfl_xor(m0, 16, 32)), shf(m1, bcol));
    C = M;
#pragma unroll
    for (int r = 0; r < 8; ++r) { V0[r] = __expf(a0[r] - M); V1[r] = __expf(a1[r] - M); }
  }

  for (int c = 0; c < NCHUNK; ++c) {
    float* buf = (c & 1) ? buf1 : buf0;
#ifdef HAVE_TDM
    if (c > 0) {                       // chunk 0 already waited above
      if (c == NCHUNK - 1) __builtin_amdgcn_s_wait_tensorcnt(0);
      else                 __builtin_amdgcn_s_wait_tensorcnt(1);
    }
#endif
    const int tl0 = (c == 0) ? 1 : 0;  // t=0 consumed by init
    for (int tl = tl0; tl < TCH; ++tl) {
      const int    t    = c * TCH + tl;
      const float* slab = buf + tl * SLABDW;
#ifndef HAVE_TDM
      copy_slab(em + embase + (size_t)t * NTAG, (float*)slab, lane);
#endif
      // Rebuild B operand from V: one ds_bpermute + cndmask per register.
      v2f Bv[6];
      {
        float s;
        s = shf(V0[2], bcol);      Bv[0].x = lo ? V0[0] : s;
        s = shf(V0[3], bcol);      Bv[0].y = lo ? V0[1] : s;
        s = shf(V0[6], bcol);      Bv[1].x = lo ? V0[4] : s;
        s = shf(V0[7], bcol);      Bv[1].y = lo ? V0[5] : s;
        s = shf(V0[0], bcol + 16); Bv[2].x = lo ? s : V0[2];
        s = shf(V0[1], bcol + 16); Bv[2].y = lo ? s : V0[3];
        s = shf(V0[4], bcol + 16); Bv[3].x = lo ? s : V0[6];
        s = shf(V0[5], bcol + 16); Bv[3].y = lo ? s : V0[7];
        s = shf(V1[2], bcol);      Bv[4].x = lo ? V1[0] : s;
        s = shf(V1[3], bcol);      Bv[4].y = lo ? V1[1] : s;
        s = shf(V1[6], bcol);      Bv[5].x = lo ? V1[4] : s;
        s = shf(V1[7], bcol);      Bv[5].y = lo ? V1[5] : s;
      }

      v8f d0 = {0.f,0.f,0.f,0.f,0.f,0.f,0.f,0.f};
      v8f d1 = {0.f,0.f,0.f,0.f,0.f,0.f,0.f,0.f};
#pragma unroll
      for (int k = 0; k < 6; ++k) {
        d0 = __builtin_amdgcn_wmma_f32_16x16x4_f32(false, A0[k], false, Bv[k],
                                                   (short)0, d0, false, false);
        d1 = __builtin_amdgcn_wmma_f32_16x16x4_f32(false, A1[k], false, Bv[k],
                                                   (short)0, d1, false, false);
      }

      // P = raw * exp(emit); renormalize by per-batch max; accumulate log C.
      float P0[8], P1[8];
#pragma unroll
      for (int r = 0; r < 8; ++r) {
        P0[r] = d0[r] * __expf(slab[off0 + r]);
        P1[r] = d1[r] * __expf(slab[off1 + r]);
      }
      float m0 = P0[0], m1 = P1[0];
#pragma unroll
      for (int r = 1; r < 8; ++r) { m0 = fmaxf(m0, P0[r]); m1 = fmaxf(m1, P1[r]); }
      const float M    = fmaxf(fmaxf(m0, __shfl_xor(m0, 16, 32)), shf(m1, bcol));
      const float rinv = __builtin_amdgcn_rcpf(M);
      const float lM   = __logf(M);
      const bool  u    = (mask[mrow + t] != 0);
#pragma unroll
      for (int r = 0; r < 8; ++r) {
        V0[r] = u ? P0[r] * rinv : V0[r];
        V1[r] = u ? P1[r] * rinv : V1[r];
      }
      C += u ? lM : 0.0f;
    }
#ifdef HAVE_TDM
    if (c + 2 < NCHUNK)                // refill the buffer just consumed
      tdm_load_chunk(em + embase + (size_t)(c + 2) * TCH * NTAG, buf);
#endif
  }

  float s0 = 0.f, s1 = 0.f;
#pragma unroll
  for (int r = 0; r < 8; ++r) { s0 += V0[r]; s1 += V1[r]; }
  const float tot = s0 + __shfl_xor(s0, 16, 32) + s1;
  if (lo) logden[grp * 16 + bcol] = C + __logf(tot);
}

// ---------------------------------------------------------------------------
__global__ __launch_bounds__(256)
void crf_gold_kernel(const float* __restrict__ em, const int* __restrict__ tags,
                     const int* __restrict__ mask, const float* __restrict__ T,
                     float* __restrict__ gold)
{
  __shared__ float Ts[NTAG * NTAG];
  __shared__ float red[256];
  const int b = blockIdx.x, tid = threadIdx.x;
  for (int i = tid; i < NTAG * NTAG; i += 256) Ts[i] = T[i];
  __syncthreads();
  float acc = 0.f;
  for (int t = tid; t < SEQ; t += 256) {
    const int idx = b * SEQ + t;
    const int tg  = tags[idx];
    const float mf = (float)mask[idx];
    acc += em[(size_t)idx * NTAG + tg] * mf;
    if (t >= 1) acc += Ts[tags[idx - 1] * NTAG + tg] * mf;
  }
  red[tid] = acc;
  __syncthreads();
  for (int s = 128; s > 0; s >>= 1) {
    if (tid < s) red[tid] += red[tid + s];
    __syncthreads();
  }
  if (tid == 0) gold[b] = red[0];
}

__global__ __launch_bounds__(256)
void crf_reduce_kernel(const float* __restrict__ logden,
                       const float* __restrict__ gold,
                       float* __restrict__ out)
{
  __shared__ float red[256];
  const int tid = threadIdx.x;
  float acc = 0.f;
  for (int i = tid; i < NBATCH; i += 256) acc += logden[i] - gold[i];
  red[tid] = acc;
  __syncthreads();
  for (int s = 128; s > 0; s >>= 1) {
    if (tid < s) red[tid] += red[tid + s];
    __syncthreads();
  }
  if (tid == 0) out[0] = red[0] * (1.0f / (float)NBATCH);
}

extern "C" void kernel_launch(void* const* d_in, const int* in_sizes, int n_in,
                              void* d_out, int out_size, void* d_ws, size_t ws_size,
                              hipStream_t stream)
{
  const float* em   = (const float*)d_in[0];
  const int*   tags = (const int*)d_in[1];
  const int*   mask = (const int*)d_in[2];
  const float* T    = (const float*)d_in[3];
  float* wsf    = (float*)d_ws;
  float* logden = wsf;            // 2048 floats
  float* gold   = wsf + NBATCH;   // 2048 floats

  crf_partition_kernel<<<NBATCH / (16 * WAVES), WAVES * 32, 0, stream>>>(em, mask, T, logden);
  crf_gold_kernel<<<NBATCH, 256, 0, stream>>>(em, tags, mask, T, gold);
  crf_reduce_kernel<<<1, 256, 0, stream>>>(logden, gold, (float*)d_out);
}